// StaticAddAttention_58626303590869
// MI455X (gfx1250) — compile-verified
//
#include <hip/hip_runtime.h>
#include <hip/hip_bf16.h>
#include <math.h>

// ---------------------------------------------------------------------------
// Problem constants (from reference): B=2, T=512, D_IN=D_MEM=1024, A=256
// ---------------------------------------------------------------------------
#define BB   2
#define TT   512
#define DIN  1024
#define AA   256
#define EPSF 1e-13f

// GEMM tiling
#define KT   32            // K-tile depth
#define PA   36            // padded LDS row pitch for A tile (conflict-free b64 reads)
#define PB   64            // LDS row pitch for B tile
#define LDS_A_FLOATS (64 * PA)          // 2304
#define LDS_B_FLOATS (KT * PB)          // 2048
#define LDS_BUF_FLOATS (LDS_A_FLOATS + LDS_B_FLOATS)

typedef __attribute__((ext_vector_type(2))) float v2f;
typedef __attribute__((ext_vector_type(8))) float v8f;

// ---------------------------------------------------------------------------
// Hardware tanh (gfx1250 V_TANH_F32) with graceful fallback
// ---------------------------------------------------------------------------
__device__ __forceinline__ float fast_tanh(float x) {
#if __has_builtin(__builtin_amdgcn_tanhf)
    return __builtin_amdgcn_tanhf(x);
#elif __has_builtin(__builtin_amdgcn_tanh_f32)
    return __builtin_amdgcn_tanh_f32(x);
#else
    return tanhf(x);
#endif
}

// ---------------------------------------------------------------------------
// CDNA5 async global->LDS copy (ASYNCcnt-tracked), per CDNA5 ISA.
// Low 32 bits of a flat pointer into LDS == LDS byte address (ISA 10.2),
// so truncating the generic pointer is the VDST.
// ---------------------------------------------------------------------------
__device__ __forceinline__ void async_ld_b128(void* ldsPtr, const float* g) {
    unsigned off = (unsigned)(uintptr_t)ldsPtr;
    unsigned long long ga = (unsigned long long)(uintptr_t)g;
    asm volatile("global_load_async_to_lds_b128 %0, %1, off"
                 :: "v"(off), "v"(ga)
                 : "memory");
}
__device__ __forceinline__ void wait_async0() {
    asm volatile("s_wait_asynccnt 0x0" ::: "memory");
}

// ---------------------------------------------------------------------------
// FP32 WMMA GEMM: C = A(MxK) * B(KxN), row-major. M,N multiples of 64,
// K multiple of 32. Block = 128 threads = 4 waves; block computes 64x64 of C,
// each wave a 16x64 strip (4 x v_wmma_f32_16x16x4_f32 accumulators).
// K-tiles of 32 are staged to LDS with async b128 copies, double-buffered.
//
// Lane layouts per CDNA5 ISA 7.12.2:
//   A 16x4 f32 : v[0]=A[m][k0+sel], v[1]=A[m][k0+sel+1], m=lane&15, sel=(lane>>4)*2
//   B 4x16 f32 : v[0]=B[k0+sel][n],  v[1]=B[k0+sel+1][n], n=lane&15
//   C 16x16 f32: vgpr v -> row v (lanes 0-15) / row v+8 (lanes 16-31), col=lane&15
// ---------------------------------------------------------------------------
__global__ __launch_bounds__(128)
void gemm_f32_wmma(const float* __restrict__ A, const float* __restrict__ B,
                   float* __restrict__ C,
                   int K, int lda, int ldb, int ldc, int tilesN64,
                   long long strideA, long long strideB, long long strideC) {
    __shared__ __align__(16) float s_lds[2][LDS_BUF_FLOATS];

    const int batch = blockIdx.y;
    const float* Ab = A + (long long)batch * strideA;
    const float* Bb = B + (long long)batch * strideB;
    float*       Cb = C + (long long)batch * strideC;

    const int rowTile = blockIdx.x / tilesN64;
    const int colTile = blockIdx.x % tilesN64;
    const int tid  = threadIdx.x;
    const int wave = tid >> 5;
    const int lane = tid & 31;
    const int lane15 = lane & 15;
    const int kSel   = (lane >> 4) << 1;          // 0 or 2

    const float* Abase = Ab + (size_t)(rowTile * 64) * lda;      // 64 rows of A
    const float* Bbase = Bb + colTile * 64;                      // 64 cols of B

    // Cooperative async stage of one K-tile into buffer `buf`.
    auto stage = [&](int buf, int k0) {
        float* sA = &s_lds[buf][0];
        float* sB = &s_lds[buf][LDS_A_FLOATS];
#pragma unroll
        for (int i = 0; i < 4; ++i) {            // A: 64 x 32, 512 float4s
            const int q = tid + i * 128;
            const int row = q >> 3, c4 = q & 7;  // 8 float4s per row
            async_ld_b128(sA + row * PA + c4 * 4,
                          Abase + (size_t)row * lda + k0 + c4 * 4);
        }
#pragma unroll
        for (int i = 0; i < 4; ++i) {            // B: 32 x 64, 512 float4s
            const int q = tid + i * 128;
            const int row = q >> 4, c4 = q & 15; // 16 float4s per row
            async_ld_b128(sB + row * PB + c4 * 4,
                          Bbase + (size_t)(k0 + row) * ldb + c4 * 4);
        }
    };

    v8f acc0 = {}, acc1 = {}, acc2 = {}, acc3 = {};

    const int nkt = K / KT;
    stage(0, 0);

    for (int kt = 0; kt < nkt; ++kt) {
        wait_async0();          // this wave's async copies for tile kt complete
        __syncthreads();        // all waves' copies done; prev compute done

        if (kt + 1 < nkt) stage((kt + 1) & 1, (kt + 1) * KT);

        const float* sA = &s_lds[kt & 1][0];
        const float* sB = &s_lds[kt & 1][LDS_A_FLOATS];
        const float* aRow = sA + (wave * 16 + lane15) * PA + kSel;

#pragma unroll
        for (int kk = 0; kk < KT; kk += 4) {
            v2f a;
            {
                const float2 t = *(const float2*)(aRow + kk);   // ds_load_b64
                a[0] = t.x; a[1] = t.y;
            }
            const float* b0 = sB + (kk + kSel) * PB + lane15;
            const float* b1 = b0 + PB;
            v2f b;
            b[0] = b0[0];  b[1] = b1[0];
            acc0 = __builtin_amdgcn_wmma_f32_16x16x4_f32(false, a, false, b, (short)0, acc0, false, false);
            b[0] = b0[16]; b[1] = b1[16];
            acc1 = __builtin_amdgcn_wmma_f32_16x16x4_f32(false, a, false, b, (short)0, acc1, false, false);
            b[0] = b0[32]; b[1] = b1[32];
            acc2 = __builtin_amdgcn_wmma_f32_16x16x4_f32(false, a, false, b, (short)0, acc2, false, false);
            b[0] = b0[48]; b[1] = b1[48];
            acc3 = __builtin_amdgcn_wmma_f32_16x16x4_f32(false, a, false, b, (short)0, acc3, false, false);
        }
        __syncthreads();        // done reading this buffer before it is refilled
    }

    // Store C: row = tile + v (+8 for upper half-lanes), col = lane&15
    const int rowC0 = rowTile * 64 + wave * 16 + ((lane >> 4) << 3);
    const int colC0 = colTile * 64 + lane15;
#pragma unroll
    for (int v = 0; v < 8; ++v) {
        float* crow = Cb + (size_t)(rowC0 + v) * ldc + colC0;
        crow[0]  = acc0[v];
        crow[16] = acc1[v];
        crow[32] = acc2[v];
        crow[48] = acc3[v];
    }
}

// ---------------------------------------------------------------------------
// wave32 / block reductions
// ---------------------------------------------------------------------------
__device__ __forceinline__ float wave_sum(float v) {
#pragma unroll
    for (int off = 16; off >= 1; off >>= 1) v += __shfl_xor(v, off, 32);
    return v;
}
__device__ __forceinline__ float wave_max(float v) {
#pragma unroll
    for (int off = 16; off >= 1; off >>= 1) v = fmaxf(v, __shfl_xor(v, off, 32));
    return v;
}

template <bool IS_MAX>
__device__ __forceinline__ float block_reduce(float v, float* red) {
    v = IS_MAX ? wave_max(v) : wave_sum(v);
    const int lane = threadIdx.x & 31;
    const int wv   = threadIdx.x >> 5;
    const int nw   = blockDim.x >> 5;
    __syncthreads();                       // protect red[] reuse
    if (lane == 0) red[wv] = v;
    __syncthreads();
    if (wv == 0) {
        float t = (lane < nw) ? red[lane] : (IS_MAX ? -3.402823466e38f : 0.0f);
        t = IS_MAX ? wave_max(t) : wave_sum(t);
        if (lane == 0) red[0] = t;
    }
    __syncthreads();
    return red[0];
}

// ---------------------------------------------------------------------------
// Attention scores: one block per (b,t). Thread j computes
//   logits[j] = sum_a w2[a] * tanh(proj_in[t,a] + proj_mem[j,a])
// then the reference's masked-softmax quirk:
//   s = softmax(logits * m) * m ; score = s / (sum(s) + eps)
// ---------------------------------------------------------------------------
__global__ __launch_bounds__(TT)
void attn_scores(const float* __restrict__ pin,   // [B*T, A]
                 const float* __restrict__ pmem,  // [B*T, A]
                 const int*   __restrict__ mask,  // [B, T]
                 const float* __restrict__ w2,    // [A]
                 float*       __restrict__ score) // [B, T, T]
{
    const int bt = blockIdx.x;         // b*T + t
    const int b  = bt >> 9;            // T = 512
    const int j  = threadIdx.x;        // 0..511

    __shared__ __align__(16) float s_pin[AA];
    __shared__ __align__(16) float s_w2[AA];
    __shared__ float s_red[16];

    if (threadIdx.x < AA) {
        s_pin[threadIdx.x] = pin[(size_t)bt * AA + threadIdx.x];
        s_w2 [threadIdx.x] = w2[threadIdx.x];
    }
    __syncthreads();

    const float* pm = pmem + ((size_t)(b * TT + j)) * AA;
    float acc = 0.0f;
#pragma unroll 4
    for (int a = 0; a < AA; a += 4) {
        const float4 m4 = *(const float4*)(pm + a);
        const float4 p4 = *(const float4*)(s_pin + a);
        const float4 w4 = *(const float4*)(s_w2 + a);
        acc = fmaf(w4.x, fast_tanh(p4.x + m4.x), acc);
        acc = fmaf(w4.y, fast_tanh(p4.y + m4.y), acc);
        acc = fmaf(w4.z, fast_tanh(p4.z + m4.z), acc);
        acc = fmaf(w4.w, fast_tanh(p4.w + m4.w), acc);
    }

    const float mj = (float)mask[b * TT + j];
    const float x  = acc * mj;                        // logits * m

    const float mx    = block_reduce<true >(x, s_red);
    const float e     = __expf(x - mx);
    const float sumE  = block_reduce<false>(e, s_red);
    const float s     = (e / sumE) * mj;              // softmax(..)*m
    const float sumS  = block_reduce<false>(s, s_red);

    score[(size_t)bt * TT + j] = s / (sumS + EPSF);
}

// ---------------------------------------------------------------------------
// out[b,t,1024+d] = input[b,t,d]  (float4 granularity)
// ---------------------------------------------------------------------------
__global__ __launch_bounds__(256)
void concat_copy(const float* __restrict__ in, float* __restrict__ out) {
    const int idx = blockIdx.x * blockDim.x + threadIdx.x;   // float4 index
    const int row = idx >> 8;                                // DIN/4 = 256 f4 per row
    const int c4  = idx & 255;
    const float4 v = ((const float4*)in)[idx];
    *(float4*)(out + (size_t)row * (2 * DIN) + DIN + 4 * c4) = v;
}

// ---------------------------------------------------------------------------
// Launch
// ---------------------------------------------------------------------------
extern "C" void kernel_launch(void* const* d_in, const int* in_sizes, int n_in,
                              void* d_out, int out_size, void* d_ws, size_t ws_size,
                              hipStream_t stream) {
    const float* input  = (const float*)d_in[0];   // [B,T,DIN]
    const float* memory = (const float*)d_in[1];   // [B,T,DIN]
    const int*   mmask  = (const int*)  d_in[2];   // [B,T]
    const float* w1     = (const float*)d_in[3];   // [2*DIN, A]
    const float* w2     = (const float*)d_in[4];   // [A]
    float*       out    = (float*)d_out;           // [B,T,2*DIN]

    char* ws = (char*)d_ws;
    float* ws_pin  = (float*)(ws);                               // 1 MB: [B*T, A]
    float* ws_pmem = (float*)(ws + (1u << 20));                  // 1 MB: [B*T, A]
    float* ws_scr  = (float*)(ws + (2u << 20));                  // 2 MB: [B,T,T]

    const int rowsP = BB * TT;            // 1024

    // 1) proj_in = input @ w1[:DIN]          (1024 x 1024 x 256)
    {
        dim3 grid((rowsP / 64) * (AA / 64), 1);
        gemm_f32_wmma<<<grid, 128, 0, stream>>>(
            input, w1, ws_pin, DIN, DIN, AA, AA, AA / 64, 0, 0, 0);
    }
    // 2) proj_mem = memory @ w1[DIN:]
    {
        dim3 grid((rowsP / 64) * (AA / 64), 1);
        gemm_f32_wmma<<<grid, 128, 0, stream>>>(
            memory, w1 + (size_t)DIN * AA, ws_pmem, DIN, DIN, AA, AA, AA / 64, 0, 0, 0);
    }
    // 3) tanh-attention logits + masked softmax -> score [B,T,T]
    {
        attn_scores<<<BB * TT, TT, 0, stream>>>(ws_pin, ws_pmem, mmask, w2, ws_scr);
    }
    // 4) context = score @ input  -> out[..., :DIN]  (batched over B, ldc = 2*DIN)
    {
        dim3 grid((TT / 64) * (DIN / 64), BB);
        gemm_f32_wmma<<<grid, 128, 0, stream>>>(
            ws_scr, input, out, TT, TT, DIN, 2 * DIN, DIN / 64,
            (long long)TT * TT, (long long)TT * DIN, (long long)TT * 2 * DIN);
    }
    // 5) out[..., DIN:] = input
    {
        const int nf4 = BB * TT * DIN / 4;      // 262144
        concat_copy<<<nf4 / 256, 256, 0, stream>>>(input, out);
    }
}